// GCNLayer_80161269613387
// MI455X (gfx1250) — compile-verified
//
#include <hip/hip_runtime.h>

#define N_NODES 100000
#define N_EDGES 1600000
#define FDIM    128
#define BN_EPS  1e-5f

typedef float v2f __attribute__((ext_vector_type(2)));
typedef float v8f __attribute__((ext_vector_type(8)));

// ---------------------------------------------------------------------------
// 1) Degree accumulation: one thread per edge, f32 atomics into deg arrays.
// ---------------------------------------------------------------------------
__global__ void gcn_deg_kernel(const int* __restrict__ src, const int* __restrict__ dst,
                               float* __restrict__ degO, float* __restrict__ degI) {
    int e = blockIdx.x * blockDim.x + threadIdx.x;
    if (e < N_EDGES) {
        atomicAdd(&degO[src[e]], 1.0f);
        atomicAdd(&degI[dst[e]], 1.0f);
    }
}

// ---------------------------------------------------------------------------
// 2) deg -> rsqrt(clip(deg,1)) in place.
// ---------------------------------------------------------------------------
__global__ void gcn_rsqrt_kernel(float* __restrict__ degO, float* __restrict__ degI) {
    int n = blockIdx.x * blockDim.x + threadIdx.x;
    if (n < N_NODES) {
        degO[n] = rsqrtf(fmaxf(degO[n], 1.0f));
        degI[n] = rsqrtf(fmaxf(degI[n], 1.0f));
    }
}

// ---------------------------------------------------------------------------
// 3) Edge gather + scatter-add: 32 lanes per edge, float4 per lane.
//    agg[dst] += feats[src] * rsqrt(out_deg[src])
// ---------------------------------------------------------------------------
__global__ __launch_bounds__(256) void gcn_scatter_kernel(
        const float* __restrict__ feats, const int* __restrict__ src,
        const int* __restrict__ dst, const float* __restrict__ rsO,
        float* __restrict__ agg) {
    int e    = blockIdx.x * 8 + (threadIdx.x >> 5);   // 8 edges per 256-thread block
    int lane = threadIdx.x & 31;
    if (e >= N_EDGES) return;
    int s = src[e];
    int d = dst[e];
    float sc = rsO[s];
    float4 v = ((const float4*)(feats + (size_t)s * FDIM))[lane];
    float* ap = agg + (size_t)d * FDIM + lane * 4;
    atomicAdd(ap + 0, v.x * sc);
    atomicAdd(ap + 1, v.y * sc);
    atomicAdd(ap + 2, v.z * sc);
    atomicAdd(ap + 3, v.w * sc);
}

// ---------------------------------------------------------------------------
// 4) Fused dual-GEMM + bias + dropout-mask, fp32 WMMA (16x16x4).
//    x = (agg*rsqrt(in_deg)) @ W + feats @ W_res + (b + b_res), then * mask.
//    Block = 256 threads = 8 waves; block owns a 16-row M-tile; wave w owns
//    the 16-column N-tile w. A tiles live in LDS (pad 132 -> no bank conflict).
// ---------------------------------------------------------------------------
__global__ __launch_bounds__(256) void gcn_gemm_kernel(
        const float* __restrict__ agg, const float* __restrict__ rsI,
        const float* __restrict__ feats,
        const float* __restrict__ W,  const float* __restrict__ b,
        const float* __restrict__ Wr, const float* __restrict__ br,
        const float* __restrict__ mask, float* __restrict__ x) {
    __shared__ float As[16][132];   // agg * rsqrt(in_deg)
    __shared__ float Fs[16][132];   // feats
    const int tid = threadIdx.x;
    const int m0  = blockIdx.x * 16;

    // Cooperative, coalesced tile load (8 floats per thread per tile).
    for (int idx = tid; idx < 16 * FDIM; idx += 256) {
        int r = idx >> 7;
        int c = idx & 127;
        int grow = m0 + r;
        As[r][c] = agg  [(size_t)grow * FDIM + c] * rsI[grow];
        Fs[r][c] = feats[(size_t)grow * FDIM + c];
    }
    __syncthreads();

    const int lane = tid & 31;
    const int wave = tid >> 5;          // N-tile index 0..7
    const int row  = lane & 15;         // A fragment M index
    const int half = lane >> 4;         // 0: K pair {0,1}, 1: K pair {2,3}
    const int kb   = half * 2;
    const int gcol = wave * 16 + (lane & 15);

    v8f acc = {0.f, 0.f, 0.f, 0.f, 0.f, 0.f, 0.f, 0.f};
    for (int k0 = 0; k0 < FDIM; k0 += 4) {
        v2f a1, b1, a2, b2;
        a1.x = As[row][k0 + kb];
        a1.y = As[row][k0 + kb + 1];
        b1.x = W[(k0 + kb)     * FDIM + gcol];
        b1.y = W[(k0 + kb + 1) * FDIM + gcol];
        acc = __builtin_amdgcn_wmma_f32_16x16x4_f32(
                  false, a1, false, b1, (short)0, acc, false, false);

        a2.x = Fs[row][k0 + kb];
        a2.y = Fs[row][k0 + kb + 1];
        b2.x = Wr[(k0 + kb)     * FDIM + gcol];
        b2.y = Wr[(k0 + kb + 1) * FDIM + gcol];
        acc = __builtin_amdgcn_wmma_f32_16x16x4_f32(
                  false, a2, false, b2, (short)0, acc, false, false);
    }

    const float bias = b[gcol] + br[gcol];
    // D layout: VGPR r -> row r (lanes 0-15) / row r+8 (lanes 16-31).
    #pragma unroll
    for (int r = 0; r < 8; ++r) {
        int grow   = m0 + r + half * 8;
        size_t off = (size_t)grow * FDIM + gcol;
        x[off] = (acc[r] + bias) * mask[off];
    }
}

// ---------------------------------------------------------------------------
// 5) BatchNorm statistics: column-per-thread partial sums, then atomics.
// ---------------------------------------------------------------------------
__global__ __launch_bounds__(128) void gcn_bnstat_kernel(
        const float* __restrict__ x, float* __restrict__ sums, float* __restrict__ sumsq) {
    int c  = threadIdx.x;                 // 0..127 (one column per thread)
    int r0 = blockIdx.x * 256;
    int r1 = r0 + 256; if (r1 > N_NODES) r1 = N_NODES;
    float s = 0.f, ss = 0.f;
    for (int r = r0; r < r1; ++r) {
        float v = x[(size_t)r * FDIM + c];
        s  += v;
        ss += v * v;
    }
    atomicAdd(&sums[c],  s);
    atomicAdd(&sumsq[c], ss);
}

__global__ void gcn_bnfin_kernel(const float* __restrict__ sums, const float* __restrict__ sumsq,
                                 const float* __restrict__ gamma, const float* __restrict__ beta,
                                 float* __restrict__ scale, float* __restrict__ shift) {
    int c = threadIdx.x;
    float inv_n = 1.0f / (float)N_NODES;
    float mu  = sums[c] * inv_n;
    float var = sumsq[c] * inv_n - mu * mu;      // biased variance
    float sc  = gamma[c] * rsqrtf(var + BN_EPS);
    scale[c] = sc;
    shift[c] = beta[c] - mu * sc;
}

// ---------------------------------------------------------------------------
// 6) In-place normalize: out = x*scale + shift (float4 vectorized).
// ---------------------------------------------------------------------------
__global__ __launch_bounds__(256) void gcn_apply_kernel(
        float* __restrict__ x, const float* __restrict__ scale, const float* __restrict__ shift) {
    size_t i  = (size_t)blockIdx.x * blockDim.x + threadIdx.x;
    size_t n4 = (size_t)N_NODES * FDIM / 4;
    if (i >= n4) return;
    int c = (int)((i * 4) & 127);
    float4 v = ((float4*)x)[i];
    v.x = v.x * scale[c + 0] + shift[c + 0];
    v.y = v.y * scale[c + 1] + shift[c + 1];
    v.z = v.z * scale[c + 2] + shift[c + 2];
    v.w = v.w * scale[c + 3] + shift[c + 3];
    ((float4*)x)[i] = v;
}

// ---------------------------------------------------------------------------
// Launcher. Input order: feats, W, b, W_res, b_res, gamma, beta, mask, src, dst
// ---------------------------------------------------------------------------
extern "C" void kernel_launch(void* const* d_in, const int* in_sizes, int n_in,
                              void* d_out, int out_size, void* d_ws, size_t ws_size,
                              hipStream_t stream) {
    const float* feats = (const float*)d_in[0];
    const float* W     = (const float*)d_in[1];
    const float* b     = (const float*)d_in[2];
    const float* Wr    = (const float*)d_in[3];
    const float* br    = (const float*)d_in[4];
    const float* gamma = (const float*)d_in[5];
    const float* beta  = (const float*)d_in[6];
    const float* mask  = (const float*)d_in[7];
    const int*   src   = (const int*)d_in[8];
    const int*   dst   = (const int*)d_in[9];
    float* out = (float*)d_out;

    // Workspace layout (floats): agg[N*F] | rsO[N] | rsI[N] | sums[F] | sumsq[F] | scale[F] | shift[F]
    float* agg   = (float*)d_ws;
    float* rsO   = agg + (size_t)N_NODES * FDIM;
    float* rsI   = rsO + N_NODES;
    float* sums  = rsI + N_NODES;
    float* sumsq = sums + FDIM;
    float* scale = sumsq + FDIM;
    float* shift = scale + FDIM;

    // Zero accumulators (agg, degrees, BN sums) — capturable async memset.
    size_t zero_bytes = ((size_t)N_NODES * FDIM + 2 * (size_t)N_NODES + 2 * FDIM) * sizeof(float);
    hipMemsetAsync(d_ws, 0, zero_bytes, stream);

    gcn_deg_kernel    <<<(N_EDGES + 255) / 256, 256, 0, stream>>>(src, dst, rsO, rsI);
    gcn_rsqrt_kernel  <<<(N_NODES + 255) / 256, 256, 0, stream>>>(rsO, rsI);
    gcn_scatter_kernel<<<N_EDGES / 8,           256, 0, stream>>>(feats, src, dst, rsO, agg);
    gcn_gemm_kernel   <<<N_NODES / 16,          256, 0, stream>>>(agg, rsI, feats, W, b, Wr, br, mask, out);
    gcn_bnstat_kernel <<<(N_NODES + 255) / 256, 128, 0, stream>>>(out, sums, sumsq);
    gcn_bnfin_kernel  <<<1,                     128, 0, stream>>>(sums, sumsq, gamma, beta, scale, shift);
    size_t n4 = (size_t)N_NODES * FDIM / 4;
    gcn_apply_kernel  <<<(unsigned)((n4 + 255) / 256), 256, 0, stream>>>(out, scale, shift);
}